// MaskLayer_56307021250625
// MI455X (gfx1250) — compile-verified
//
#include <hip/hip_runtime.h>
#include <stdint.h>

// MaskLayer: per-channel spatial argmax + L1-distance mask multiply.
// x: [1,128,128,2048] f32 (NHWC, channel fastest). Memory-bound:
// ~384 MB HBM traffic => ~12-17 us floor at 23.3 TB/s. No matmul structure,
// so no WMMA; CDNA5 features used: async global->LDS staging (ASYNCcnt),
// global_atomic_max_u64 argmax in L2, NT loads/stores, global_prefetch.

#define N_IMG     128
#define N_CH      2048
#define N_SPATIAL (N_IMG * N_IMG)      // 16384
#define TAU_F     3.0517578125e-05f    // 0.5/16384 == 2^-15 (exact)
#define INV_N_B   0.03125f             // BETA/n == 4/128 (exact)

typedef float vfloat4 __attribute__((ext_vector_type(4)));
typedef int   v4i     __attribute__((vector_size(4 * sizeof(int))));

// Monotone float -> uint mapping (no NaNs expected from randn input).
__device__ __forceinline__ uint32_t order_f32(float f) {
    uint32_t u = __float_as_uint(f);
    return u ^ ((uint32_t)((int32_t)u >> 31) | 0x80000000u);
}

// ---------------------------------------------------------------------------
// Kernel 0: init packed argmax slots. Any real (ord<<32)|~idx is > 0.
// ---------------------------------------------------------------------------
__global__ void mask_init_best(unsigned long long* __restrict__ best) {
    int i = blockIdx.x * blockDim.x + threadIdx.x;
    if (i < N_CH) best[i] = 0ull;
}

// ---------------------------------------------------------------------------
// Kernel 1: per-channel argmax. Grid (2, 128), block 256.
// Block covers 1024 channels x 128 spatial rows. Thread owns 4 channels.
// Packed u64 = (ordered_val << 32) | ~spatial_idx  -> atomic max gives
// "largest value, smallest index on ties" == jnp.argmax semantics.
// ---------------------------------------------------------------------------
#define P1_ROWS   128
#define P1_CHUNK  8
#define P1_NCHUNK (P1_ROWS / P1_CHUNK)

__global__ __launch_bounds__(256) void mask_pass1_argmax(
    const float* __restrict__ x, unsigned long long* __restrict__ best)
{
    const int t     = threadIdx.x;
    const int cbase = blockIdx.x * 1024;
    const int c0    = cbase + t * 4;
    const int s0    = blockIdx.y * P1_ROWS;

    uint32_t bval[4] = {0u, 0u, 0u, 0u};
    int      bidx[4] = {0, 0, 0, 0};

#if defined(__AMDGCN__) && defined(__has_builtin)
#if __has_builtin(__builtin_amdgcn_global_load_async_to_lds_b128) && \
    __has_builtin(__builtin_amdgcn_s_wait_asynccnt)
#define MASK_ASYNC_PATH 1
#endif
#endif

#ifdef MASK_ASYNC_PATH
    // Double-buffered async global->LDS staging: 2 x [8 rows x 1024 f32] = 64 KB.
    __shared__ float tile[2][P1_CHUNK * 1024];
    const int wave = t >> 5;
    const int lane = t & 31;

    auto issue = [&](int chunk, int buf) {
        // One spatial row per wave; 8 x B128 per wave fills 4 KB row segment.
        const int sp = s0 + chunk * P1_CHUNK + wave;
        const float* g = x + (size_t)sp * N_CH + cbase;
        float* l = &tile[buf][wave * 1024];
#pragma unroll
        for (int q = 0; q < 8; ++q) {
            const float* gp = g + q * 128 + lane * 4;
            float*       lp = l + q * 128 + lane * 4;
            // Param types per hipcc diagnostic: (AS1 v4i*, AS3 v4i*, imm, imm).
            __builtin_amdgcn_global_load_async_to_lds_b128(
                (__attribute__((address_space(1))) v4i*)(uintptr_t)gp,
                (__attribute__((address_space(3))) v4i*)(uint32_t)(uintptr_t)lp,
                0, 0);
        }
    };

    issue(0, 0);
    for (int k = 0; k < P1_NCHUNK; ++k) {
        if (k + 1 < P1_NCHUNK) {
            issue(k + 1, (k + 1) & 1);
            __builtin_amdgcn_s_wait_asynccnt(8);  // chunk k's 8 done (in-order)
        } else {
            __builtin_amdgcn_s_wait_asynccnt(0);
        }
        __syncthreads();
        const float* l = &tile[k & 1][0];
#pragma unroll
        for (int r = 0; r < P1_CHUNK; ++r) {
            vfloat4 v = *(const vfloat4*)(l + r * 1024 + t * 4);
            const int sp = s0 + k * P1_CHUNK + r;
#pragma unroll
            for (int q = 0; q < 4; ++q) {
                uint32_t ou = order_f32(v[q]);
                if (ou > bval[q]) { bval[q] = ou; bidx[q] = sp; }
            }
        }
        __syncthreads();   // release buffer k&1 for refill
    }
#else
    // Fallback: direct strided B128 streaming + prefetch.
    const vfloat4* p = (const vfloat4*)(x + (size_t)s0 * N_CH + c0);
    for (int s = 0; s < P1_ROWS; ++s) {
        if (s + 8 < P1_ROWS) __builtin_prefetch(p + (size_t)(s + 8) * 512, 0, 3);
        vfloat4 v = p[(size_t)s * 512];   // row stride = 2048 f32 = 512 vfloat4
        const int sp = s0 + s;
#pragma unroll
        for (int q = 0; q < 4; ++q) {
            uint32_t ou = order_f32(v[q]);
            if (ou > bval[q]) { bval[q] = ou; bidx[q] = sp; }
        }
    }
#endif

#pragma unroll
    for (int q = 0; q < 4; ++q) {
        unsigned long long pk =
            ((unsigned long long)bval[q] << 32) | (uint32_t)(~(uint32_t)bidx[q]);
        atomicMax(&best[c0 + q], pk);
    }
}

// ---------------------------------------------------------------------------
// Kernel 2: apply mask. Grid (2, 256), block 256; thread owns 4 channels,
// streams 64 rows. NT loads (last use of x) + NT stores (never re-read) keep
// L2 clean. Mask math is bit-exact vs reference (power-of-two scalings).
// ---------------------------------------------------------------------------
#define P2_ROWS 64

__global__ __launch_bounds__(256) void mask_pass2_apply(
    const float* __restrict__ x, const unsigned long long* __restrict__ best,
    float* __restrict__ out)
{
    const int t  = threadIdx.x;
    const int c0 = blockIdx.x * 1024 + t * 4;
    const int s0 = blockIdx.y * P2_ROWS;

    int row[4], col[4];
#pragma unroll
    for (int q = 0; q < 4; ++q) {
        uint32_t idx = ~(uint32_t)best[c0 + q];   // low 32 bits hold ~argmax
        row[q] = (int)(idx >> 7);
        col[q] = (int)(idx & 127u);
    }

    const vfloat4* src = (const vfloat4*)(x   + (size_t)s0 * N_CH + c0);
    vfloat4*       dst = (vfloat4*)      (out + (size_t)s0 * N_CH + c0);

    for (int s = 0; s < P2_ROWS; ++s) {
        if (s + 8 < P2_ROWS) __builtin_prefetch(src + (size_t)(s + 8) * 512, 0, 0);
        const int sp = s0 + s;
        const int ii = sp >> 7;
        const int jj = sp & 127;
        vfloat4 v = __builtin_nontemporal_load(src + (size_t)s * 512);
        vfloat4 r;
#pragma unroll
        for (int q = 0; q < 4; ++q) {
            int dist = __builtin_abs(ii - row[q]) + __builtin_abs(jj - col[q]);
            float m  = TAU_F * fmaxf(1.0f - INV_N_B * (float)dist, -1.0f);
            r[q] = v[q] * m;
        }
        __builtin_nontemporal_store(r, dst + (size_t)s * 512);
    }
}

// ---------------------------------------------------------------------------
extern "C" void kernel_launch(void* const* d_in, const int* in_sizes, int n_in,
                              void* d_out, int out_size, void* d_ws, size_t ws_size,
                              hipStream_t stream) {
    const float* x = (const float*)d_in[0];
    float* out = (float*)d_out;
    // Workspace: 2048 packed u64 argmax slots (16 KB).
    unsigned long long* best = (unsigned long long*)d_ws;

    mask_init_best<<<dim3(8), dim3(256), 0, stream>>>(best);
    mask_pass1_argmax<<<dim3(2, 128), dim3(256), 0, stream>>>(x, best);
    mask_pass2_apply <<<dim3(2, 256), dim3(256), 0, stream>>>(x, best, out);
}